// Graph_Layer_gatHeadFullStep_36507222016272
// MI455X (gfx1250) — compile-verified
//
#include <hip/hip_runtime.h>
#include <hip/hip_bf16.h>

typedef float v2f __attribute__((ext_vector_type(2)));
typedef float v8f __attribute__((ext_vector_type(8)));

#define D        256
#define DEG      8
#define LVLS     8
#define NEG_SLOPE 0.2f
#define LN_EPS    1e-5f

// -------------------------------------------------------------------------
// H = A @ W^T :  H[m,n] = sum_k A[m,k] * W[n,k]
// A: [M,256] row-major, W: [256,256] row-major, H: [M,256]
// One wave computes a 16x64 output strip via 4 parallel V_WMMA_F32_16X16X4_F32
// accumulator chains. Fragment layouts per CDNA5 ISA 7.12.2 (32-bit):
//   A 16x4 : lane<16 -> M=lane, {v0,v1}={K,K+1}; lane>=16 -> {K+2,K+3}
//   B 4x16 : lane<16 -> N=lane, {v0,v1}={K,K+1}; lane>=16 -> {K+2,K+3}
//   C/D    : VGPR r -> row M = r + (lane>=16 ? 8 : 0), col N = lane&15
// Launched with block(32,8); grid sized exactly (no divergence: EXEC all-1s).
// -------------------------------------------------------------------------
__global__ void gemm_xwT_wmma(const float* __restrict__ A,
                              const float* __restrict__ Wm,
                              float* __restrict__ H, int M) {
  const int lane  = threadIdx.x;                       // 0..31
  const int wave  = blockIdx.x * blockDim.y + threadIdx.y;
  const int mtiles = M >> 4;                           // tiles along M
  const int mt = wave & (mtiles - 1);
  const int ng = wave / mtiles;                        // 0..3 : n-group of 64
  const int hi  = (lane >= 16) ? 1 : 0;
  const int l15 = lane & 15;
  const int m0 = mt << 4;
  const int n0 = ng << 6;
  const int koff = hi ? 2 : 0;

  const float* __restrict__ Arow = A  + (size_t)(m0 + l15) * D;
  const float* __restrict__ W0   = Wm + (size_t)(n0 + l15) * D;      // B col n -> W row n
  const float* __restrict__ W1   = W0 + 16 * D;
  const float* __restrict__ W2   = W0 + 32 * D;
  const float* __restrict__ W3   = W0 + 48 * D;

  v8f acc0 = {}, acc1 = {}, acc2 = {}, acc3 = {};
  #pragma unroll 4
  for (int k = 0; k < D; k += 4) {
    const int ka = k + koff;
    v2f a  = *(const v2f*)(Arow + ka);
    v2f b0 = *(const v2f*)(W0 + ka);
    v2f b1 = *(const v2f*)(W1 + ka);
    v2f b2 = *(const v2f*)(W2 + ka);
    v2f b3 = *(const v2f*)(W3 + ka);
    acc0 = __builtin_amdgcn_wmma_f32_16x16x4_f32(false, a, false, b0, (short)0, acc0, false, false);
    acc1 = __builtin_amdgcn_wmma_f32_16x16x4_f32(false, a, false, b1, (short)0, acc1, false, false);
    acc2 = __builtin_amdgcn_wmma_f32_16x16x4_f32(false, a, false, b2, (short)0, acc2, false, false);
    acc3 = __builtin_amdgcn_wmma_f32_16x16x4_f32(false, a, false, b3, (short)0, acc3, false, false);
  }

  const int rh = hi ? 8 : 0;
  #pragma unroll
  for (int r = 0; r < 8; ++r) {
    float* hr = H + (size_t)(m0 + r + rh) * D + l15;
    hr[n0]      = acc0[r];
    hr[n0 + 16] = acc1[r];
    hr[n0 + 32] = acc2[r];
    hr[n0 + 48] = acc3[r];
  }
}

// out[r] = rows[r,:] . vec   (one wave per row)
__global__ void rowdot_kernel(const float* __restrict__ rows,
                              const float* __restrict__ vec,
                              float* __restrict__ out, int nrows) {
  const int gw   = (blockIdx.x * blockDim.x + threadIdx.x) >> 5;
  const int lane = threadIdx.x & 31;
  if (gw >= nrows) return;
  const float* r = rows + (size_t)gw * D;
  float s = 0.f;
  #pragma unroll
  for (int c = 0; c < D; c += 32) s += r[c + lane] * vec[c + lane];
  #pragma unroll
  for (int off = 16; off > 0; off >>= 1) s += __shfl_xor(s, off, 32);
  if (lane == 0) out[gw] = s;
}

// vdst[j] = sum_i W[i,j]*att_dst[i];  cdst = bias . vdst   (single block, 256 thr)
__global__ void vdst_kernel(const float* __restrict__ Wm,
                            const float* __restrict__ att_dst,
                            const float* __restrict__ bias,
                            float* __restrict__ vdst, float* __restrict__ cdst) {
  const int j = threadIdx.x;
  float s = 0.f;
  for (int i = 0; i < D; ++i) s += Wm[(size_t)i * D + j] * att_dst[i];
  vdst[j] = s;
  __shared__ float red[D];
  red[j] = s * bias[j];
  __syncthreads();
  for (int off = 128; off > 0; off >>= 1) {
    if (j < off) red[j] += red[j + off];
    __syncthreads();
  }
  if (j == 0) cdst[0] = red[0];
}

// One wave per dst node: softmax over its DEG contiguous edges, gather-accumulate
// alpha * h[src].  Each lane owns 8 channels (two float4s).
__global__ void gat_agg_kernel(const int* __restrict__ src_l, const int* __restrict__ dst_l,
                               const float* __restrict__ h,   const float* __restrict__ a_src,
                               const float* __restrict__ a_dst, const float* __restrict__ cdst,
                               const float* __restrict__ bias, float* __restrict__ outbuf,
                               int level, int per, int use_adst_vec) {
  const int w    = blockIdx.x * (blockDim.x >> 5) + (threadIdx.x >> 5);
  const int lane = threadIdx.x & 31;
  if (w >= per) return;
  const int e0    = w * DEG;
  const int dstv  = dst_l[e0];
  const int j     = dstv - level * per;
  const int sbase = (level - 1) * per;
  const float adst = use_adst_vec ? a_dst[j] : cdst[0];

  int   sidx[DEG];
  float ev[DEG];
  float m = -1e30f;
  #pragma unroll
  for (int k = 0; k < DEG; ++k) {
    sidx[k] = src_l[e0 + k] - sbase;
    float v = a_src[sidx[k]] + adst;
    ev[k] = (v > 0.f) ? v : NEG_SLOPE * v;
    m = fmaxf(m, ev[k]);
  }
  float ssum = 0.f;
  #pragma unroll
  for (int k = 0; k < DEG; ++k) { ev[k] = expf(ev[k] - m); ssum += ev[k]; }
  const float inv = 1.f / ssum;

  float4 acc0 = {0.f, 0.f, 0.f, 0.f}, acc1 = {0.f, 0.f, 0.f, 0.f};
  #pragma unroll
  for (int k = 0; k < DEG; ++k) {
    const float al = ev[k] * inv;
    const float* hr = h + (size_t)sidx[k] * D + lane * 8;
    float4 f0 = *(const float4*)(hr);
    float4 f1 = *(const float4*)(hr + 4);
    acc0.x += al * f0.x; acc0.y += al * f0.y; acc0.z += al * f0.z; acc0.w += al * f0.w;
    acc1.x += al * f1.x; acc1.y += al * f1.y; acc1.z += al * f1.z; acc1.w += al * f1.w;
  }
  const float* bc = bias + lane * 8;
  float4 r0 = { acc0.x + bc[0], acc0.y + bc[1], acc0.z + bc[2], acc0.w + bc[3] };
  float4 r1 = { acc1.x + bc[4], acc1.y + bc[5], acc1.z + bc[6], acc1.w + bc[7] };
  float* o = outbuf + (size_t)j * D + lane * 8;
  *(float4*)(o)     = r0;
  *(float4*)(o + 4) = r1;
}

__global__ void zero2_kernel(float* sums) {
  if (threadIdx.x < 2) sums[threadIdx.x] = 0.f;
}

// sum and sum-of-squares of y = x + (level7 ? top : bias)
__global__ void reduce_y_kernel(const float* __restrict__ x, const float* __restrict__ top,
                                const float* __restrict__ bias, float* __restrict__ sums,
                                int N, int per) {
  const int total = N * D;
  const int l7 = (LVLS - 1) * per;
  float s = 0.f, s2 = 0.f;
  for (int idx = blockIdx.x * blockDim.x + threadIdx.x; idx < total;
       idx += gridDim.x * blockDim.x) {
    const int n = idx >> 8, c = idx & (D - 1);
    float v = x[idx] + ((n >= l7) ? top[(size_t)(n - l7) * D + c] : bias[c]);
    s += v; s2 += v * v;
  }
  __shared__ float sh1[256], sh2[256];
  const int tid = threadIdx.x;
  sh1[tid] = s; sh2[tid] = s2;
  __syncthreads();
  for (int off = 128; off > 0; off >>= 1) {
    if (tid < off) { sh1[tid] += sh1[tid + off]; sh2[tid] += sh2[tid + off]; }
    __syncthreads();
  }
  if (tid == 0) { atomicAdd(&sums[0], sh1[0]); atomicAdd(&sums[1], sh2[0]); }
}

__global__ void norm_out_kernel(const float* __restrict__ x, const float* __restrict__ top,
                                const float* __restrict__ bias, const float* __restrict__ gamma,
                                const float* __restrict__ beta, const float* __restrict__ sums,
                                float* __restrict__ out, int N, int per) {
  const int total = N * D;
  const int l7 = (LVLS - 1) * per;
  const float invc = 1.f / (float)total;
  const float mean = sums[0] * invc;
  const float var  = sums[1] * invc - mean * mean;
  const float rstd = 1.f / sqrtf(var + LN_EPS);
  for (int idx = blockIdx.x * blockDim.x + threadIdx.x; idx < total;
       idx += gridDim.x * blockDim.x) {
    const int n = idx >> 8, c = idx & (D - 1);
    float v = x[idx] + ((n >= l7) ? top[(size_t)(n - l7) * D + c] : bias[c]);
    out[idx] = (v - mean) * rstd * gamma[c] + beta[c];
  }
}

__global__ void copy4_kernel(const float* __restrict__ in, float* __restrict__ out, int n4) {
  const float4* __restrict__ i4 = (const float4*)in;
  float4* __restrict__ o4 = (float4*)out;
  for (int i = blockIdx.x * blockDim.x + threadIdx.x; i < n4;
       i += gridDim.x * blockDim.x)
    o4[i] = i4[i];
}

extern "C" void kernel_launch(void* const* d_in, const int* in_sizes, int n_in,
                              void* d_out, int out_size, void* d_ws, size_t ws_size,
                              hipStream_t stream) {
  const float* x        = (const float*)d_in[0];
  const int*   ei       = (const int*)d_in[1];
  const float* edgeattr = (const float*)d_in[2];
  const float* Wm       = (const float*)d_in[5];
  const float* att_src  = (const float*)d_in[6];
  const float* att_dst  = (const float*)d_in[7];
  const float* bias     = (const float*)d_in[8];
  const float* gamma    = (const float*)d_in[9];
  const float* beta     = (const float*)d_in[10];

  const int N    = in_sizes[0] / D;   // 65536
  const int E    = in_sizes[1] / 2;   // 458752
  const int per  = N / LVLS;          // 8192
  const int Eper = per * DEG;         // 65536

  const int* srcp = ei;
  const int* dstp = ei + E;

  float* ws      = (float*)d_ws;
  float* hbuf    = ws;                              // per*D
  float* ping    = hbuf   + (size_t)per * D;        // per*D
  float* pong    = ping   + (size_t)per * D;        // per*D
  float* a_src_b = pong   + (size_t)per * D;        // per
  float* a_dst_b = a_src_b + per;                   // per
  float* vdst    = a_dst_b + per;                   // D
  float* cdst    = vdst + D;                        // 1
  float* sums    = cdst + 1;                        // 2

  // Precompute v = W^T att_dst, cdst = bias.v, and a_dst for level-1 targets.
  vdst_kernel<<<1, D, 0, stream>>>(Wm, att_dst, bias, vdst, cdst);
  rowdot_kernel<<<(per * 32 + 255) / 256, 256, 0, stream>>>(
      x + (size_t)per * D, vdst, a_dst_b, per);

  const int gemm_blocks = ((per / 16) * 4) / 8;     // waves / (8 waves per block)
  const float* curA = x;                            // level-0 rows of x
  float* outb = ping;
  for (int l = 1; l < LVLS; ++l) {
    gemm_xwT_wmma<<<gemm_blocks, dim3(32, 8), 0, stream>>>(curA, Wm, hbuf, per);
    rowdot_kernel<<<(per * 32 + 255) / 256, 256, 0, stream>>>(hbuf, att_src, a_src_b, per);
    gat_agg_kernel<<<per / 8, 256, 0, stream>>>(
        srcp + (size_t)(l - 1) * Eper, dstp + (size_t)(l - 1) * Eper,
        hbuf, a_src_b, a_dst_b, cdst, bias, outb, l, per, (l == 1) ? 1 : 0);
    curA = outb;
    outb = (outb == ping) ? pong : ping;
  }
  const float* top = curA;                          // level-7 GAT outputs

  float* outp = (float*)d_out;
  zero2_kernel<<<1, 32, 0, stream>>>(sums);
  reduce_y_kernel<<<2048, 256, 0, stream>>>(x, top, bias, sums, N, per);
  norm_out_kernel<<<2048, 256, 0, stream>>>(x, top, bias, gamma, beta, sums, outp, N, per);

  // edge_attr passthrough into output tail
  copy4_kernel<<<8192, 256, 0, stream>>>(edgeattr, outp + (size_t)N * D, (E * D) / 4);
}